// GAT_19911468384538
// MI455X (gfx1250) — compile-verified
//
#include <hip/hip_runtime.h>
#include <math.h>

typedef float v2f __attribute__((ext_vector_type(2)));
typedef float v8f __attribute__((ext_vector_type(8)));

// ---- monotonic float<->uint encoding for atomicMax-based segment max ----
__device__ __forceinline__ unsigned fenc(float f) {
  unsigned u = __float_as_uint(f);
  return (u >> 31) ? ~u : (u | 0x80000000u);
}
__device__ __forceinline__ float fdec(unsigned e) {
  return (e >> 31) ? __uint_as_float(e & 0x7fffffffu) : __uint_as_float(~e);
}

__device__ __forceinline__ v8f wmma_f32(v2f a, v2f b, v8f c) {
  return __builtin_amdgcn_wmma_f32_16x16x4_f32(
      /*neg_a=*/false, a, /*neg_b=*/false, b,
      /*c_mod=*/(short)0, c, /*reuse_a=*/false, /*reuse_b=*/false);
}

// =====================================================================
// GEMM: C[M,N] = A[M,K] @ B[K,N], exact f32 via V_WMMA_F32_16X16X4_F32.
// One wave computes a 64x16 output strip (4 stacked 16x16 tiles): the
// B fragment is loaded once per k-step and reused by 4 WMMAs.
// Branch-free inner loop: B column clamped for N-tail tiles (WMMA
// column j of D depends only on column j of B; stores are masked),
// A rows clamped for M-tail subtiles (stores masked per subtile).
// Fragment layouts per ISA 7.12.2:
//   A 16x4 f32: lanes 0-15 row=lane, v{0,1}=K{0,1}; lanes 16-31 K{2,3}.
//   B 4x16:     v0 lanes0-15 K=0 / lanes16-31 K=2; v1 K=1 / K=3.
//   C/D:        VGPR i -> lanes 0-15 row i, lanes 16-31 row i+8.
// =====================================================================
__global__ void __launch_bounds__(256) gemm_wmma_f32_64x16(
    const float* __restrict__ A, const float* __restrict__ B,
    float* __restrict__ C, int M, int K, int N) {
  int lane   = threadIdx.x & 31;
  int wave   = threadIdx.x >> 5;
  int tilesN = (N + 15) >> 4;
  int tilesM = (M + 63) >> 6;
  int tile   = blockIdx.x * (blockDim.x >> 5) + wave;
  if (tile >= tilesM * tilesN) return;
  int tm = (tile / tilesN) << 6;
  int tn = (tile % tilesN) << 4;

  int row  = lane & 15;
  int koff = (lane >> 4) << 1;              // 0 or 2
  int col  = tn + (lane & 15);
  bool colOK = (col < N);
  int colC = colOK ? col : (N - 1);         // clamp; stores masked below

  // A pointers for the 4 stacked M-subtiles (rows clamped for M tail)
  const float* pa0;
  const float* pa1;
  const float* pa2;
  const float* pa3;
  {
    int r0 = tm + 0  + row; if (r0 >= M) r0 = M - 1;
    int r1 = tm + 16 + row; if (r1 >= M) r1 = M - 1;
    int r2 = tm + 32 + row; if (r2 >= M) r2 = M - 1;
    int r3 = tm + 48 + row; if (r3 >= M) r3 = M - 1;
    pa0 = A + (size_t)r0 * K + koff;
    pa1 = A + (size_t)r1 * K + koff;
    pa2 = A + (size_t)r2 * K + koff;
    pa3 = A + (size_t)r3 * K + koff;
  }
  const float* pb0 = B + (size_t)koff * N + colC;   // row k+koff
  const float* pb1 = pb0 + N;                       // row k+koff+1
  const size_t bstep = (size_t)4 * N;

  v8f acc0 = {}, acc1 = {}, acc2 = {}, acc3 = {};
  for (int k = 0; k < K; k += 4) {
    v2f b;
    b.x = *pb0;
    b.y = *pb1;
    v2f a0 = *(const v2f*)pa0;
    v2f a1 = *(const v2f*)pa1;
    v2f a2 = *(const v2f*)pa2;
    v2f a3 = *(const v2f*)pa3;
    acc0 = wmma_f32(a0, b, acc0);
    acc1 = wmma_f32(a1, b, acc1);
    acc2 = wmma_f32(a2, b, acc2);
    acc3 = wmma_f32(a3, b, acc3);
    pa0 += 4; pa1 += 4; pa2 += 4; pa3 += 4;
    pb0 += bstep; pb1 += bstep;
  }

  if (colOK) {
    int rsub = (lane >> 4) << 3;            // 0 or 8 within each subtile
#pragma unroll
    for (int j = 0; j < 4; ++j) {
      int base = tm + 16 * j;
      if (base >= M) break;                  // M % 16 == 0 -> whole subtile
      v8f acc = (j == 0) ? acc0 : (j == 1) ? acc1 : (j == 2) ? acc2 : acc3;
#pragma unroll
      for (int i = 0; i < 8; ++i)
        C[(size_t)(base + rsub + i) * N + col] = acc[i];
    }
  }
}

// =====================================================================
// Edge pass 1: per (edge, head): logit = sum_c att[h,c]*lrelu(xl+xr);
// store logit, segment-max into menc[dst,h] (encoded u32 atomicMax).
// =====================================================================
__global__ void __launch_bounds__(256) edge_logit_max(
    const float* __restrict__ xl, const float* __restrict__ xr,
    const float* __restrict__ att, const int* __restrict__ src,
    const int* __restrict__ dst, int E, long long total, int H, int Cc,
    float* __restrict__ logits, unsigned* __restrict__ menc) {
  long long t = (long long)blockIdx.x * blockDim.x + threadIdx.x;
  if (t >= total) return;
  int e = (int)(t / H);
  int h = (int)(t % H);
  int s, d;
  if (e < E) { s = src[e]; d = dst[e]; } else { s = e - E; d = s; }
  int F = H * Cc;
  const float* pl = xl + (size_t)s * F + h * Cc;
  const float* pr = xr + (size_t)d * F + h * Cc;
  const float* pa = att + h * Cc;
  float acc = 0.f;
  for (int c = 0; c < Cc; ++c) {
    float v = pl[c] + pr[c];
    v = v > 0.f ? v : 0.2f * v;              // leaky_relu, slope 0.2
    acc = fmaf(pa[c], v, acc);
  }
  logits[t] = acc;
  atomicMax(&menc[(size_t)d * H + h], fenc(acc));
}

// =====================================================================
// Edge pass 2: denom[dst,h] += exp(logit - max)
// =====================================================================
__global__ void __launch_bounds__(256) edge_denom_k(
    const int* __restrict__ dst, int E, long long total, int H,
    const float* __restrict__ logits, const unsigned* __restrict__ menc,
    float* __restrict__ denom) {
  long long t = (long long)blockIdx.x * blockDim.x + threadIdx.x;
  if (t >= total) return;
  int e = (int)(t / H);
  int h = (int)(t % H);
  int d = (e < E) ? dst[e] : (e - E);
  float m = fdec(menc[(size_t)d * H + h]);
  atomicAdd(&denom[(size_t)d * H + h], expf(logits[t] - m));
}

// =====================================================================
// Edge pass 3: agg[dst, h*C + c] += alpha * xl[src, h*C + c]
// =====================================================================
__global__ void __launch_bounds__(256) edge_scatter(
    const float* __restrict__ xl, const int* __restrict__ src,
    const int* __restrict__ dst, int E, long long total, int H, int Cc,
    const float* __restrict__ logits, const unsigned* __restrict__ menc,
    const float* __restrict__ denom, float* __restrict__ agg) {
  long long t = (long long)blockIdx.x * blockDim.x + threadIdx.x;
  if (t >= total) return;
  int e = (int)(t / H);
  int h = (int)(t % H);
  int s, d;
  if (e < E) { s = src[e]; d = dst[e]; } else { s = e - E; d = s; }
  size_t nh = (size_t)d * H + h;
  float m = fdec(menc[nh]);
  float w = expf(logits[t] - m) / (denom[nh] + 1e-16f);
  int F = H * Cc;
  const float* pl = xl + (size_t)s * F + h * Cc;
  float* po = agg + (size_t)d * F + h * Cc;
  for (int c = 0; c < Cc; ++c)
    atomicAdd(&po[c], w * pl[c]);
}

// =====================================================================
// bias + ELU  (between layers)
// =====================================================================
__global__ void __launch_bounds__(256) bias_elu(
    const float* __restrict__ agg, const float* __restrict__ bias,
    float* __restrict__ out, long long total, int F) {
  long long i = (long long)blockIdx.x * blockDim.x + threadIdx.x;
  if (i >= total) return;
  float v = agg[i] + bias[(int)(i % F)];
  out[i] = v > 0.f ? v : (expf(v) - 1.f);
}

// =====================================================================
// bias + log_softmax over F=47 classes. One wave per node (wave32:
// lanes cover f and f+32), shuffle reductions for max and sum-exp.
// =====================================================================
__global__ void __launch_bounds__(256) bias_log_softmax(
    const float* __restrict__ agg, const float* __restrict__ bias,
    float* __restrict__ out, int n, int F) {
  int wv   = (int)(((long long)blockIdx.x * blockDim.x + threadIdx.x) >> 5);
  int lane = threadIdx.x & 31;
  if (wv >= n) return;
  const float* row = agg + (size_t)wv * F;
  float v0 = (lane < F)      ? row[lane]      + bias[lane]      : -INFINITY;
  float v1 = (lane + 32 < F) ? row[lane + 32] + bias[lane + 32] : -INFINITY;
  float mx = fmaxf(v0, v1);
#pragma unroll
  for (int o = 16; o; o >>= 1) mx = fmaxf(mx, __shfl_xor(mx, o, 32));
  float sm = ((lane < F)      ? expf(v0 - mx) : 0.f) +
             ((lane + 32 < F) ? expf(v1 - mx) : 0.f);
#pragma unroll
  for (int o = 16; o; o >>= 1) sm += __shfl_xor(sm, o, 32);
  float lse = mx + logf(sm);
  if (lane < F)      out[(size_t)wv * F + lane]      = v0 - lse;
  if (lane + 32 < F) out[(size_t)wv * F + lane + 32] = v1 - lse;
}

// =====================================================================
extern "C" void kernel_launch(void* const* d_in, const int* in_sizes, int n_in,
                              void* d_out, int out_size, void* d_ws, size_t ws_size,
                              hipStream_t stream) {
  (void)n_in; (void)out_size; (void)ws_size;
  const float* x   = (const float*)d_in[0];
  const int*   ei  = (const int*)d_in[1];
  const float* Wl0 = (const float*)d_in[2];
  const float* Wr0 = (const float*)d_in[3];
  const float* a0  = (const float*)d_in[4];
  const float* b0  = (const float*)d_in[5];
  const float* Wl1 = (const float*)d_in[6];
  const float* Wr1 = (const float*)d_in[7];
  const float* a1  = (const float*)d_in[8];
  const float* b1  = (const float*)d_in[9];
  const float* Wl2 = (const float*)d_in[10];
  const float* Wr2 = (const float*)d_in[11];
  const float* a2  = (const float*)d_in[12];
  const float* b2  = (const float*)d_in[13];

  const int NFEAT = 128, HID = 256, H = 8, C = 32, NCLASS = 47;
  int n  = in_sizes[0] / NFEAT;
  int E  = in_sizes[1] / 2;
  int ET = E + n;                 // with self-loops
  const int* src = ei;
  const int* dst = ei + E;

  // workspace layout (floats)
  size_t nF = (size_t)n * HID;
  float*    xl     = (float*)d_ws;
  float*    xr     = xl + nF;
  float*    logits = xr + nF;
  unsigned* menc   = (unsigned*)(logits + (size_t)ET * H);
  float*    denom  = (float*)(menc + (size_t)n * H);
  float*    agg    = denom + (size_t)n * H;
  float*    hbuf   = agg + nF;

  auto cdiv = [](long long a, long long b) { return (int)((a + b - 1) / b); };

  auto gemm = [&](const float* A, const float* Bm, float* Cm,
                  int M, int K, int Ncols) {
    int tiles = ((M + 63) >> 6) * ((Ncols + 15) >> 4);
    gemm_wmma_f32_64x16<<<cdiv(tiles, 8), 256, 0, stream>>>(A, Bm, Cm, M, K,
                                                            Ncols);
  };

  auto edge_phase = [&](const float* att, int Hh, int Cc) {
    int F = Hh * Cc;
    hipMemsetAsync(menc,  0, (size_t)n * Hh * sizeof(unsigned), stream);
    hipMemsetAsync(denom, 0, (size_t)n * Hh * sizeof(float), stream);
    hipMemsetAsync(agg,   0, (size_t)n * F * sizeof(float), stream);
    long long tot = (long long)ET * Hh;
    int blocks = cdiv(tot, 256);
    edge_logit_max<<<blocks, 256, 0, stream>>>(xl, xr, att, src, dst, E, tot,
                                               Hh, Cc, logits, menc);
    edge_denom_k<<<blocks, 256, 0, stream>>>(dst, E, tot, Hh, logits, menc,
                                             denom);
    edge_scatter<<<blocks, 256, 0, stream>>>(xl, src, dst, E, tot, Hh, Cc,
                                             logits, menc, denom, agg);
  };

  // ---------------- layer 0: 128 -> 256 (8 heads x 32) ----------------
  gemm(x, Wl0, xl, n, NFEAT, HID);
  gemm(x, Wr0, xr, n, NFEAT, HID);
  edge_phase(a0, H, C);
  bias_elu<<<cdiv((long long)n * HID, 256), 256, 0, stream>>>(
      agg, b0, hbuf, (long long)n * HID, HID);

  // ---------------- layer 1: 256 -> 256 (8 heads x 32) ----------------
  gemm(hbuf, Wl1, xl, n, HID, HID);
  gemm(hbuf, Wr1, xr, n, HID, HID);
  edge_phase(a1, H, C);
  bias_elu<<<cdiv((long long)n * HID, 256), 256, 0, stream>>>(
      agg, b1, hbuf, (long long)n * HID, HID);

  // ---------------- layer 2: 256 -> 47 (1 head) + log_softmax ---------
  gemm(hbuf, Wl2, xl, n, HID, NCLASS);
  gemm(hbuf, Wr2, xr, n, HID, NCLASS);
  edge_phase(a2, 1, NCLASS);
  bias_log_softmax<<<cdiv((long long)n * 32, 256), 256, 0, stream>>>(
      agg, b2, (float*)d_out, n, NCLASS);
}